// Experts_36034775614075
// MI455X (gfx1250) — compile-verified
//
#include <hip/hip_runtime.h>

typedef float v8f __attribute__((ext_vector_type(8)));
typedef float v2f __attribute__((ext_vector_type(2)));
typedef int   v4i128 __attribute__((vector_size(16)));   // matches builtin's param print

typedef __attribute__((address_space(1))) void*   as1_vp;
typedef __attribute__((address_space(3))) void*   as3_vp;
typedef __attribute__((address_space(1))) v4i128* as1_p128;
typedef __attribute__((address_space(3))) v4i128* as3_p128;

// Problem constants (match reference)
#define TT      16384          // tokens = SEQ*BATCH
#define DM      1024           // model dim
#define NE      8              // experts
#define CAP     2048           // capacity = T/E
#define NBLK    256            // routing blocks (64 tokens each)

// d_out layout: out[T*D] | counts[E] | probs_sum[E] | n_dropped[1] | pmax[T]
#define COUNTS_OFF  (TT * DM)
#define PSUM_OFF    (COUNTS_OFF + NE)
#define ND_OFF      (PSUM_OFF + NE)
#define PMAX_OFF    (ND_OFF + 1)

// GEMM tiling
#define BM   128               // block tile M
#define BN   256               // block tile N
#define BK   32                // K slab
#define LDST 36                // LDS row stride (16B aligned, bank-clean)
#define A_ELEMS (BM * LDST)    // 4608 floats per A buffer
#define B_ELEMS (BN * LDST)    // 9216 floats per B buffer
#define GEMM_LDS_BYTES ((2 * A_ELEMS + 2 * B_ELEMS) * 4)   // 110592 B

// CDNA5 async global->LDS path (guarded; falls back to load+ds_store)
#if defined(__AMDGCN__) && __has_builtin(__builtin_amdgcn_global_load_async_to_lds_b128) && __has_builtin(__builtin_amdgcn_s_wait_asynccnt)
#define ASYNC_LDS 1
#else
#define ASYNC_LDS 0
#endif

__device__ __forceinline__ void cp16_g2l(const float* __restrict__ g, float* l) {
#if ASYNC_LDS
    __builtin_amdgcn_global_load_async_to_lds_b128(
        (as1_p128)(as1_vp)g,
        (as3_p128)(as3_vp)l, 0, 0);
#else
    *(float4*)l = *(const float4*)g;
#endif
}

__device__ __forceinline__ void async_fence() {
#if ASYNC_LDS
    __builtin_amdgcn_s_wait_asynccnt(0);
#endif
}

// ---------------------------------------------------------------- init
__global__ void moe_init(float* __restrict__ out) {
    int i = threadIdx.x;
    if (i < 2 * NE + 1) out[COUNTS_OFF + i] = 0.0f;
}

// ---------------------------------------------------------------- routing
__global__ __launch_bounds__(256) void moe_route(
    const float* __restrict__ x, const float* __restrict__ wsw,
    const float* __restrict__ bsw, float* __restrict__ out,
    int* __restrict__ ridx, int* __restrict__ bcnt)
{
    __shared__ float ws[NE * DM];   // 32 KB
    __shared__ int hist[NE];
    const int tid = threadIdx.x;
    for (int i = tid; i < NE * DM; i += 256) ws[i] = wsw[i];
    if (tid < NE) hist[tid] = 0;
    __syncthreads();

    const int lane = tid & 31;
    const int wid  = tid >> 5;

    for (int i = 0; i < 8; ++i) {
        const int t = blockIdx.x * 64 + wid * 8 + i;
        float acc[NE] = {};
        const float* xr = x + (size_t)t * DM;
        for (int d = lane; d < DM; d += 32) {
            const float xv = xr[d];
#pragma unroll
            for (int e = 0; e < NE; ++e) acc[e] += xv * ws[e * DM + d];
        }
#pragma unroll
        for (int off = 16; off > 0; off >>= 1) {
#pragma unroll
            for (int e = 0; e < NE; ++e) acc[e] += __shfl_down(acc[e], off, 32);
        }
        if (lane == 0) {
            float lg[NE], p[NE];
            float m = -1e30f;
#pragma unroll
            for (int e = 0; e < NE; ++e) { lg[e] = acc[e] + bsw[e]; m = fmaxf(m, lg[e]); }
            float s = 0.0f;
#pragma unroll
            for (int e = 0; e < NE; ++e) { p[e] = expf(lg[e] - m); s += p[e]; }
            const float inv = 1.0f / s;
            int best = 0; float bv = lg[0];
#pragma unroll
            for (int e = 1; e < NE; ++e) if (lg[e] > bv) { bv = lg[e]; best = e; }
            ridx[t] = best;
            out[PMAX_OFF + t] = p[best] * inv;
#pragma unroll
            for (int e = 0; e < NE; ++e) atomicAdd(&out[PSUM_OFF + e], p[e] * inv);
            atomicAdd(&hist[best], 1);
        }
    }
    __syncthreads();
    if (tid < NE) {
        bcnt[blockIdx.x * NE + tid] = hist[tid];
        atomicAdd(&out[COUNTS_OFF + tid], (float)hist[tid]);
    }
}

// ---------------------------------------------------------------- scan
__global__ void moe_scan(int* __restrict__ bcnt, float* __restrict__ out) {
    __shared__ int tot[NE];
    const int tid = threadIdx.x;
    if (tid < NE) {
        int run = 0;
        for (int b = 0; b < NBLK; ++b) {
            const int c = bcnt[b * NE + tid];
            bcnt[b * NE + tid] = run;
            run += c;
        }
        tot[tid] = run;
    }
    __syncthreads();
    if (tid == 0) {
        int nd = 0;
#pragma unroll
        for (int e = 0; e < NE; ++e) nd += max(0, tot[e] - CAP);
        out[ND_OFF] = (float)nd;
    }
}

// ---------------------------------------------------------------- dispatch
__global__ __launch_bounds__(256) void moe_dispatch(
    const float* __restrict__ x, const int* __restrict__ ridx,
    int* __restrict__ pos, const int* __restrict__ boffs,
    float* __restrict__ buf)
{
    const int b = blockIdx.x, tid = threadIdx.x;
    if (tid < NE) {
        int run = boffs[b * NE + tid];
        for (int i = 0; i < 64; ++i) {
            const int t = b * 64 + i;
            if (ridx[t] == tid) { pos[t] = run; run++; }
        }
    }
    __syncthreads();
    for (int i = 0; i < 64; ++i) {
        const int t = b * 64 + i;
        const int p = pos[t];
        if (p < CAP) {
            const int e = ridx[t];
            const float4* src = (const float4*)(x + (size_t)t * DM);
            float4* dst = (float4*)(buf + ((size_t)e * CAP + p) * DM);
            dst[tid] = src[tid];
        }
    }
}

// ---------------------------------------------------------------- expert GEMM
// Per expert: dst[c,f] = sum_d src[c,d]*W[f,d] + bias[f].
// Block tile 128x256, K slab 32, double-buffered LDS with async global->LDS.
// 8 waves (2M x 4N), each wave computes 64x64 via 4x4 accumulators of 16x16.
// Both A and B live row-major [row][k] with stride 36 — the WMMA A and B
// fragments use the *same* per-lane pattern (two consecutive K at fixed M/N),
// so no transpose is needed anywhere.
__global__ __launch_bounds__(256) void moe_gemm(
    const float* __restrict__ src, const float* __restrict__ W,
    const float* __restrict__ bias, float* __restrict__ dst)
{
    extern __shared__ float smem[];   // [As0|As1|Bs0|Bs1]
    float* const AsBuf = smem;
    float* const BsBuf = smem + 2 * A_ELEMS;

    const int tid = threadIdx.x;
    const int e  = blockIdx.z;
    const int m0 = blockIdx.y * BM;
    const int n0 = blockIdx.x * BN;
    const int lane  = tid & 31, wid = tid >> 5;
    const int waveM = wid >> 2, waveN = wid & 3;   // 2x4 wave grid
    const int lhalf = lane & 15, hi = lane >> 4;

    const float* Abase = src + ((size_t)e * CAP + m0) * DM;
    const float* Wbase = W   + ((size_t)e * DM  + n0) * DM;

    const v8f zero = {0.f, 0.f, 0.f, 0.f, 0.f, 0.f, 0.f, 0.f};
    v8f acc[4][4];
#pragma unroll
    for (int mi = 0; mi < 4; ++mi)
#pragma unroll
        for (int ni = 0; ni < 4; ++ni) acc[mi][ni] = zero;

    const int r0 = tid >> 3;          // 0..31
    const int c4 = (tid & 7) * 4;     // 0..28

    // stage one K slab (A: 128 rows, B: 256 rows) into buffer `bi`
    auto stage = [&](int bi, int k0) {
        float* As = AsBuf + bi * A_ELEMS;
        float* Bs = BsBuf + bi * B_ELEMS;
#pragma unroll
        for (int pass = 0; pass < BM / 32; ++pass) {
            const int r = pass * 32 + r0;
            cp16_g2l(Abase + (size_t)r * DM + k0 + c4, As + r * LDST + c4);
        }
#pragma unroll
        for (int pass = 0; pass < BN / 32; ++pass) {
            const int n = pass * 32 + r0;
            cp16_g2l(Wbase + (size_t)n * DM + k0 + c4, Bs + n * LDST + c4);
        }
    };

    int cur = 0;
    stage(cur, 0);
    async_fence();
    __syncthreads();

    for (int k0 = 0; k0 < DM; k0 += BK) {
        if (k0 + BK < DM) stage(cur ^ 1, k0 + BK);   // prefetch next slab

        const float* As = AsBuf + cur * A_ELEMS;
        const float* Bs = BsBuf + cur * B_ELEMS;
#pragma unroll
        for (int kk = 0; kk < BK; kk += 4) {
            v2f a[4], b[4];
            const int kb = kk + hi * 2;
#pragma unroll
            for (int mi = 0; mi < 4; ++mi) {
                const int row = waveM * 64 + mi * 16 + lhalf;
                a[mi][0] = As[row * LDST + kb];
                a[mi][1] = As[row * LDST + kb + 1];
            }
#pragma unroll
            for (int ni = 0; ni < 4; ++ni) {
                const int col = waveN * 64 + ni * 16 + lhalf;
                b[ni][0] = Bs[col * LDST + kb];
                b[ni][1] = Bs[col * LDST + kb + 1];
            }
#pragma unroll
            for (int mi = 0; mi < 4; ++mi)
#pragma unroll
                for (int ni = 0; ni < 4; ++ni)
                    acc[mi][ni] = __builtin_amdgcn_wmma_f32_16x16x4_f32(
                        false, a[mi], false, b[ni],
                        (short)0, acc[mi][ni], false, false);
        }
        async_fence();
        __syncthreads();
        cur ^= 1;
    }

    // Epilogue: C 16x16 layout — VGPR i: lanes 0-15 -> row i, lanes 16-31 -> row i+8
#pragma unroll
    for (int ni = 0; ni < 4; ++ni) {
        const int coll = waveN * 64 + ni * 16 + lhalf;
        const float bv = bias[(size_t)e * DM + n0 + coll];
#pragma unroll
        for (int mi = 0; mi < 4; ++mi) {
#pragma unroll
            for (int i = 0; i < 8; ++i) {
                const int row = m0 + waveM * 64 + mi * 16 + hi * 8 + i;
                dst[((size_t)e * CAP + row) * DM + n0 + coll] = acc[mi][ni][i] + bv;
            }
        }
    }
}

// ---------------------------------------------------------------- combine
__global__ __launch_bounds__(256) void moe_combine(
    const float* __restrict__ x, const int* __restrict__ ridx,
    const int* __restrict__ pos, const float* __restrict__ buf,
    float* __restrict__ out)
{
    const int t = blockIdx.x, tid = threadIdx.x;
    const int p = pos[t];
    const int e = ridx[t];
    const float pm = out[PMAX_OFF + t];
    const float4* srcr = (p < CAP)
        ? (const float4*)(buf + ((size_t)e * CAP + p) * DM)
        : (const float4*)(x + (size_t)t * DM);
    float4 v = srcr[tid];
    v.x *= pm; v.y *= pm; v.z *= pm; v.w *= pm;
    ((float4*)(out + (size_t)t * DM))[tid] = v;
}

// ---------------------------------------------------------------- launch
extern "C" void kernel_launch(void* const* d_in, const int* in_sizes, int n_in,
                              void* d_out, int out_size, void* d_ws, size_t ws_size,
                              hipStream_t stream) {
    const float* x    = (const float*)d_in[0];
    const float* wsw  = (const float*)d_in[1];
    const float* bsw  = (const float*)d_in[2];
    const float* wexp = (const float*)d_in[3];
    const float* bexp = (const float*)d_in[4];
    float* out = (float*)d_out;
    char*  ws  = (char*)d_ws;

    int* ridx = (int*)ws;             // [T]
    int* pos  = ridx + TT;            // [T]
    int* bcnt = pos + TT;             // [NBLK*E]
    float* buf = (float*)(ws + (size_t)(1 << 20));                             // 64 MB
    float* h1  = (float*)(ws + (size_t)(1 << 20) + (size_t)NE * CAP * DM * 4); // 64 MB

    moe_init<<<1, 32, 0, stream>>>(out);
    moe_route<<<NBLK, 256, 0, stream>>>(x, wsw, bsw, out, ridx, bcnt);
    moe_scan<<<1, 32, 0, stream>>>(bcnt, out);
    moe_dispatch<<<NBLK, 256, 0, stream>>>(x, ridx, pos, bcnt, buf);

    dim3 g(DM / BN, CAP / BM, NE);    // (ntile, mtile, expert)
    moe_gemm<<<g, 256, GEMM_LDS_BYTES, stream>>>(buf, wexp, bexp, h1);          // layer 0
    moe_gemm<<<g, 256, GEMM_LDS_BYTES, stream>>>(h1, wexp + (size_t)NE * DM * DM,
                                                 bexp + NE * DM, buf);          // layer 1

    moe_combine<<<TT, 256, 0, stream>>>(x, ridx, pos, buf, out);
}